// GraphemeColourSynaesthesiaNet_35888746726080
// MI455X (gfx1250) — compile-verified
//
#include <hip/hip_runtime.h>

// GraphemeColourSynaesthesiaNet — MI455X (gfx1250) implementation.
// Memory-bound sequential recurrence; K(64MB)+W(32MB) resident in 192MB L2.
// K mat-vecs run on the CDNA5 tensor path via V_WMMA_F32_16X16X4_F32.

#define Mdim 4096
#define Ndim 2048
#define NCHUNK 16   // row-chunks for the energy column reduction

typedef __attribute__((ext_vector_type(2))) float v2f;
typedef __attribute__((ext_vector_type(8))) float v8f;

__device__ __forceinline__ float sigmoidf(float z) {
    return 1.0f / (1.0f + __expf(-z));
}

// ---------------------------------------------------------------------------
// Zero the state vector and losses[0] (reference sets losses[0] = 0).
__global__ void k_init(float* __restrict__ s, float* __restrict__ losses) {
    int i = blockIdx.x * blockDim.x + threadIdx.x;
    if (i < Mdim) s[i] = 0.0f;
    if (i == 0)  losses[0] = 0.0f;
}

// ---------------------------------------------------------------------------
// wx[i] = dot(W[i,:], x)  — loop invariant, computed once. One wave per row.
__global__ void k_wx(const float* __restrict__ W, const float* __restrict__ x,
                     float* __restrict__ wx) {
    const int lane = threadIdx.x & 31;
    const int wv   = threadIdx.x >> 5;
    const int row  = blockIdx.x * (blockDim.x >> 5) + wv;
    if (row >= Mdim) return;
    const float4* Wr = (const float4*)(W + (size_t)row * Ndim);
    const float4* xr = (const float4*)x;
    float sum = 0.0f;
    for (int j = lane; j < Ndim / 4; j += 32) {
        float4 a = Wr[j], b = xr[j];
        sum += a.x * b.x + a.y * b.y + a.z * b.z + a.w * b.w;
    }
    #pragma unroll
    for (int off = 16; off > 0; off >>= 1) sum += __shfl_down(sum, off, 32);
    if (lane == 0) wx[row] = sum;
}

// ---------------------------------------------------------------------------
// dst = sigmoid(wx + K @ src), optionally g2 = (dst*(1-dst))^2.
// Block = 16 waves, one 16-row tile of K per block; waves split k (4096) into
// 256-wide chunks. Each k-step is one V_WMMA_F32_16X16X4_F32:
//   A frag (16x4 f32): lanes 0-15 -> (row, k0..k1), lanes 16-31 -> (row, k2..k3)
//   B frag (4x16 f32): EVERY column n carries src[k] (broadcast — no masking,
//     no EXEC divergence). Then every column of D equals K_tile @ src_chunk;
//     we read column 0 via lanes 0 / 16.
// src (16 KB) is staged in LDS so the B fetch is a broadcast ds_load_b64.
__global__ void __launch_bounds__(512)
k_matvec_sig(const float* __restrict__ K, const float* __restrict__ src,
             const float* __restrict__ wx, float* __restrict__ dst,
             float* __restrict__ g2) {
    __shared__ float sv[Mdim];        // staged src vector (16 KB)
    __shared__ float red[16 * 16];    // cross-wave reduction tile

    // Cooperative stage of src into LDS: 512 threads x 2 float4 = 4096 floats.
    {
        const float4* s4 = (const float4*)src;
        float4* sv4 = (float4*)sv;
        #pragma unroll
        for (int i = threadIdx.x; i < Mdim / 4; i += 512) sv4[i] = s4[i];
    }
    __syncthreads();

    const int lane  = threadIdx.x & 31;
    const int wv    = threadIdx.x >> 5;
    const int m0    = blockIdx.x << 4;            // 16-row tile base
    const int row   = m0 + (lane & 15);
    const int khalf = (lane >> 4) << 1;           // 0 for lanes 0-15, 2 for 16-31
    const float* Krow = K + (size_t)row * Mdim;

    v8f acc0 = {};
    v8f acc1 = {};
    int kb = wv * (Mdim / 16);                    // 256-wide k chunk per wave
    const int kend = kb + (Mdim / 16);
    for (; kb < kend; kb += 8) {
        {   // k = kb .. kb+3
            const float2 av = *(const float2*)(Krow + kb + khalf);
            const float2 bv = *(const float2*)(sv + kb + khalf);   // LDS broadcast
            v2f a; a.x = av.x; a.y = av.y;
            v2f b; b.x = bv.x; b.y = bv.y;
            acc0 = __builtin_amdgcn_wmma_f32_16x16x4_f32(
                false, a, false, b, (short)0, acc0, false, false);
        }
        {   // k = kb+4 .. kb+7 (second accumulator breaks the RAW chain)
            const float2 av = *(const float2*)(Krow + kb + 4 + khalf);
            const float2 bv = *(const float2*)(sv + kb + 4 + khalf);
            v2f a; a.x = av.x; a.y = av.y;
            v2f b; b.x = bv.x; b.y = bv.y;
            acc1 = __builtin_amdgcn_wmma_f32_16x16x4_f32(
                false, a, false, b, (short)0, acc1, false, false);
        }
    }
    v8f acc = acc0 + acc1;

    // Columns of D are identical; lane 0 holds rows m0..m0+7, lane 16 holds
    // rows m0+8..m0+15 (VGPR v = row m0 + v (+8)).
    if (lane == 0) {
        #pragma unroll
        for (int v = 0; v < 8; ++v) red[wv * 16 + v] = acc[v];
    } else if (lane == 16) {
        #pragma unroll
        for (int v = 0; v < 8; ++v) red[wv * 16 + 8 + v] = acc[v];
    }
    __syncthreads();

    if (threadIdx.x < 16) {
        float sum = 0.0f;
        #pragma unroll
        for (int w2 = 0; w2 < 16; ++w2) sum += red[w2 * 16 + threadIdx.x];
        float sgm = sigmoidf(wx[m0 + threadIdx.x] + sum);
        dst[m0 + threadIdx.x] = sgm;
        if (g2) {
            float gp = sgm * (1.0f - sgm);
            g2[m0 + threadIdx.x] = gp * gp;
        }
    }
}

// ---------------------------------------------------------------------------
// ypart[c][j] = sum_{i in chunk c} g2[i] * W[i][j]^2   (deterministic partials,
// fully coalesced: consecutive threads read consecutive j).
__global__ void k_colacc(const float* __restrict__ W, const float* __restrict__ g2,
                         float* __restrict__ ypart) {
    const int j  = blockIdx.x * blockDim.x + threadIdx.x;
    const int c  = blockIdx.y;
    const int i0 = c * (Mdim / NCHUNK);
    float sum = 0.0f;
    #pragma unroll 4
    for (int i = i0; i < i0 + (Mdim / NCHUNK); ++i) {
        float wvv = W[(size_t)i * Ndim + j];
        sum += g2[i] * wvv * wvv;
    }
    ypart[(size_t)c * Ndim + j] = sum;
}

// ---------------------------------------------------------------------------
// losses[t] = -0.5 * sum_j log( sum_c ypart[c][j] ) — single-block, deterministic.
__global__ void k_lossred(const float* __restrict__ ypart, float* __restrict__ out_t) {
    __shared__ float sh[256];
    float ls = 0.0f;
    for (int j = threadIdx.x; j < Ndim; j += 256) {
        float y = 0.0f;
        #pragma unroll
        for (int c = 0; c < NCHUNK; ++c) y += ypart[(size_t)c * Ndim + j];
        ls += logf(y);
    }
    sh[threadIdx.x] = ls;
    __syncthreads();
    for (int s = 128; s > 0; s >>= 1) {
        if (threadIdx.x < s) sh[threadIdx.x] += sh[threadIdx.x + s];
        __syncthreads();
    }
    if (threadIdx.x == 0) *out_t = -0.5f * sh[0];
}

// ---------------------------------------------------------------------------
extern "C" void kernel_launch(void* const* d_in, const int* in_sizes, int n_in,
                              void* d_out, int out_size, void* d_ws, size_t ws_size,
                              hipStream_t stream) {
    const float* x = (const float*)d_in[0];   // 2048
    const float* W = (const float*)d_in[1];   // 4096 x 2048
    const float* K = (const float*)d_in[2];   // 4096 x 4096
    float* losses  = (float*)d_out;           // T floats

    float* s     = (float*)d_ws;              // [Mdim]
    float* sa    = s  + Mdim;                 // [Mdim]
    float* g2    = sa + Mdim;                 // [Mdim]
    float* wx    = g2 + Mdim;                 // [Mdim]
    float* ypart = wx + Mdim;                 // [NCHUNK * Ndim]

    const int T = out_size;                   // int(max_iter/DT) == 500, host-known

    k_init<<<(Mdim + 255) / 256, 256, 0, stream>>>(s, losses);
    k_wx<<<Mdim / 8, 256, 0, stream>>>(W, x, wx);

    for (int t = 1; t < T; ++t) {
        // s_a = sigmoid(Wx + K s)
        k_matvec_sig<<<Mdim / 16, 512, 0, stream>>>(K, s, wx, sa, nullptr);
        // s = sigmoid(Wx + K s_a), g2 = (s(1-s))^2
        k_matvec_sig<<<Mdim / 16, 512, 0, stream>>>(K, sa, wx, s, g2);
        // y_j = sum_i g2_i W_ij^2 (partials), then E = -0.5 sum log y
        k_colacc<<<dim3(Ndim / 256, NCHUNK), 256, 0, stream>>>(W, g2, ypart);
        k_lossred<<<1, 256, 0, stream>>>(ypart, losses + t);
    }
}